// GATModel_13365938225825
// MI455X (gfx1250) — compile-verified
//
#include <hip/hip_runtime.h>
#include <hip/hip_bf16.h>
#include <math.h>

typedef __attribute__((ext_vector_type(2))) float v2f;
typedef __attribute__((ext_vector_type(8))) float v8f;

#define NEG_SLOPE 0.2f

// -------- float atomic max via signed-max / unsigned-min bit trick --------
__device__ __forceinline__ void atomicMaxF32(float* addr, float val) {
  if (val >= 0.0f) {
    atomicMax((int*)addr, __float_as_int(val));
  } else {
    atomicMin((unsigned int*)addr, __float_as_uint(val));
  }
}

// -------- generic fill --------
__global__ void fill_kernel(float* __restrict__ p, float v, int n) {
  int t = blockIdx.x * blockDim.x + threadIdx.x;
  if (t < n) p[t] = v;
}

// -------- C[M,N] = A[M,K] @ B[K,N], fp32, row-major. M%16==0, N%64==0, K%4==0.
// One wave32 computes a 16x64 strip (4 WMMA tiles) via V_WMMA_F32_16X16X4_F32,
// reusing one A fragment across 4 WMMAs per K-step.
// A frag (16x4): lanes 0-15 hold rows M=lane, K=k0..k0+1 in VGPR0/1;
//                lanes 16-31 hold rows M=lane-16, K=k0+2..k0+3.
// B frag (4x16): mirrored — lanes 0-15: K=k0,k0+1 ; lanes 16-31: K=k0+2,k0+3; N=lane&15.
// C/D (16x16): VGPR i = row m0 + i + (lane>=16 ? 8 : 0), col n0 + (lane&15).
__global__ void gemm_wmma_f32(const float* __restrict__ A, const float* __restrict__ B,
                              float* __restrict__ C, int M, int K, int N) {
  int wave = (blockIdx.x * blockDim.x + threadIdx.x) >> 5;
  int lane = threadIdx.x & 31;
  int stripsN = N >> 6;                      // strips of 64 columns
  int strips  = (M >> 4) * stripsN;
  if (wave >= strips) return;                // wave-uniform: EXEC stays all-1s
  int tm = wave / stripsN;
  int tn = wave % stripsN;
  int m0 = tm << 4, n0 = tn << 6;
  int lr = lane & 15;
  int kh = (lane >> 4) << 1;                 // 0 or 2
  const float* arow = A + (size_t)(m0 + lr) * K + kh;
  const float* bcol = B + (size_t)kh * N + n0 + lr;
  v8f acc0 = {}, acc1 = {}, acc2 = {}, acc3 = {};
#pragma unroll 4
  for (int k0 = 0; k0 < K; k0 += 4) {
    v2f a;
    a.x = arow[k0];
    a.y = arow[k0 + 1];
    const float* brow0 = bcol + (size_t)k0 * N;
    const float* brow1 = bcol + (size_t)(k0 + 1) * N;
    v2f b0, b1, b2, b3;
    b0.x = brow0[0];  b0.y = brow1[0];
    b1.x = brow0[16]; b1.y = brow1[16];
    b2.x = brow0[32]; b2.y = brow1[32];
    b3.x = brow0[48]; b3.y = brow1[48];
    acc0 = __builtin_amdgcn_wmma_f32_16x16x4_f32(false, a, false, b0, (short)0, acc0, false, false);
    acc1 = __builtin_amdgcn_wmma_f32_16x16x4_f32(false, a, false, b1, (short)0, acc1, false, false);
    acc2 = __builtin_amdgcn_wmma_f32_16x16x4_f32(false, a, false, b2, (short)0, acc2, false, false);
    acc3 = __builtin_amdgcn_wmma_f32_16x16x4_f32(false, a, false, b3, (short)0, acc3, false, false);
  }
  float* crow = C + (size_t)(m0 + ((lane >> 4) << 3)) * N + n0 + lr;
#pragma unroll
  for (int i = 0; i < 8; ++i) {
    crow[(size_t)i * N +  0] = acc0[i];
    crow[(size_t)i * N + 16] = acc1[i];
    crow[(size_t)i * N + 32] = acc2[i];
    crow[(size_t)i * N + 48] = acc3[i];
  }
}

// -------- el[n,h] = sum_d feat[n,h,d]*al[h,d] ; er likewise --------
__global__ void node_attn_coef(const float* __restrict__ feat, const float* __restrict__ al,
                               const float* __restrict__ ar, float* __restrict__ el,
                               float* __restrict__ er, int NH, int H, int D) {
  int t = blockIdx.x * blockDim.x + threadIdx.x;  // t = n*H + h
  if (t >= NH) return;
  int h = t % H;
  const float* f  = feat + (size_t)t * D;
  const float* wl = al + h * D;
  const float* wr = ar + h * D;
  float sl = 0.f, sr = 0.f;
  for (int d = 0; d < D; ++d) {
    float v = f[d];
    sl += v * wl[d];
    sr += v * wr[d];
  }
  el[t] = sl;
  er[t] = sr;
}

// -------- pass 1: logits = leakyrelu(el[src]+er[dst]); per-dst max --------
__global__ void edge_logits_max(const int* __restrict__ src, const int* __restrict__ dst,
                                const float* __restrict__ el, const float* __restrict__ er,
                                float* __restrict__ elog, float* __restrict__ m,
                                int EH, int H) {
  int t = blockIdx.x * blockDim.x + threadIdx.x;  // t = e*H + h
  if (t >= EH) return;
  int e = t / H;
  int h = t - e * H;
  float v = el[src[e] * H + h] + er[dst[e] * H + h];
  v = (v > 0.0f) ? v : NEG_SLOPE * v;
  elog[t] = v;
  atomicMaxF32(&m[dst[e] * H + h], v);
}

// -------- pass 2: ex = exp(logit - m[dst]); per-dst sum --------
__global__ void edge_exp_sum(const int* __restrict__ dst, float* __restrict__ exv,
                             const float* __restrict__ m, float* __restrict__ s,
                             int EH, int H) {
  int t = blockIdx.x * blockDim.x + threadIdx.x;
  if (t >= EH) return;
  int e = t / H;
  int h = t - e * H;
  int d = dst[e];
  float v = expf(exv[t] - m[d * H + h]);
  exv[t] = v;
  atomicAdd(&s[d * H + h], v);
}

// -------- pass 3: out[dst] += feat[src] * (ex / s[dst]) ; one wave per edge ----
#define AGG_LOOKAHEAD 64
__global__ void edge_aggregate(const int* __restrict__ src, const int* __restrict__ dst,
                               const float* __restrict__ feat, const float* __restrict__ exv,
                               const float* __restrict__ s, float* __restrict__ out,
                               int E, int H, int D, int F) {
  int wave = (blockIdx.x * blockDim.x + threadIdx.x) >> 5;
  int lane = threadIdx.x & 31;
  if (wave >= E) return;
  int sn = src[wave];
  int dn = dst[wave];
  int vpl  = F >> 5;              // values per lane: 8 (F=256) or 2 (F=64)
  int base = lane * vpl;          // contiguous slice; h constant per lane
  // L2-latency-hiding prefetch of a future edge's source row (gfx1250 path)
  if (wave + AGG_LOOKAHEAD < E) {
    int psn = src[wave + AGG_LOOKAHEAD];
    __builtin_prefetch(feat + (size_t)psn * F + base, 0, 1);
  }
  int h = base / D;
  float a = exv[(size_t)wave * H + h] / s[(size_t)dn * H + h];
  const float* fs = feat + (size_t)sn * F + base;
  float*       od = out  + (size_t)dn * F + base;
  if (vpl == 8) {
    float4 x0 = ((const float4*)fs)[0];
    float4 x1 = ((const float4*)fs)[1];
    atomicAdd(&od[0], x0.x * a); atomicAdd(&od[1], x0.y * a);
    atomicAdd(&od[2], x0.z * a); atomicAdd(&od[3], x0.w * a);
    atomicAdd(&od[4], x1.x * a); atomicAdd(&od[5], x1.y * a);
    atomicAdd(&od[6], x1.z * a); atomicAdd(&od[7], x1.w * a);
  } else {
    float2 x = ((const float2*)fs)[0];
    atomicAdd(&od[0], x.x * a);
    atomicAdd(&od[1], x.y * a);
  }
}

// -------- bias (+ optional relu), in place --------
__global__ void bias_act(float* __restrict__ x, const float* __restrict__ b,
                         int NF, int F, int relu) {
  int t = blockIdx.x * blockDim.x + threadIdx.x;
  if (t >= NF) return;
  float v = x[t] + b[t % F];
  if (relu) v = (v > 0.0f) ? v : 0.0f;
  x[t] = v;
}

// ======================== host orchestration ========================
static void run_gat_layer(const float* hin, int Fin, const float* W, const float* al,
                          const float* ar, const float* bias, int H, int D,
                          const int* src, const int* dst, int N, int E,
                          float* feat, float* el, float* er, float* m, float* s,
                          float* ex, float* out, int relu, hipStream_t stream) {
  const int Fout = H * D;
  // GEMM: feat = hin @ W   (one wave per 16x64 strip)
  {
    int strips = (N / 16) * (Fout / 64);
    int blocks = (strips + 7) / 8;                // 8 waves per 256-thread block
    gemm_wmma_f32<<<blocks, 256, 0, stream>>>(hin, W, feat, N, Fin, Fout);
  }
  const int NH = N * H;
  const int NF = N * Fout;
  const int EH = E * H;
  node_attn_coef<<<(NH + 255) / 256, 256, 0, stream>>>(feat, al, ar, el, er, NH, H, D);
  fill_kernel<<<(NH + 255) / 256, 256, 0, stream>>>(m, -INFINITY, NH);
  fill_kernel<<<(NH + 255) / 256, 256, 0, stream>>>(s, 0.0f, NH);
  fill_kernel<<<(NF + 255) / 256, 256, 0, stream>>>(out, 0.0f, NF);
  edge_logits_max<<<(EH + 255) / 256, 256, 0, stream>>>(src, dst, el, er, ex, m, EH, H);
  edge_exp_sum<<<(EH + 255) / 256, 256, 0, stream>>>(dst, ex, m, s, EH, H);
  edge_aggregate<<<(E * 32 + 255) / 256, 256, 0, stream>>>(src, dst, feat, ex, s, out,
                                                           E, H, D, Fout);
  bias_act<<<(NF + 255) / 256, 256, 0, stream>>>(out, bias, NF, Fout, relu);
}

extern "C" void kernel_launch(void* const* d_in, const int* in_sizes, int n_in,
                              void* d_out, int out_size, void* d_ws, size_t ws_size,
                              hipStream_t stream) {
  const int IN = 128, H = 4, D = 64, OUT = 64;
  const int N = in_sizes[0] / IN;
  const int E = in_sizes[1];

  const float* features = (const float*)d_in[0];
  const int*   src      = (const int*)d_in[1];
  const int*   dst      = (const int*)d_in[2];
  const float* W1  = (const float*)d_in[3];
  const float* al1 = (const float*)d_in[4];
  const float* ar1 = (const float*)d_in[5];
  const float* b1  = (const float*)d_in[6];
  const float* W2  = (const float*)d_in[7];
  const float* al2 = (const float*)d_in[8];
  const float* ar2 = (const float*)d_in[9];
  const float* b2  = (const float*)d_in[10];
  const float* W3  = (const float*)d_in[11];
  const float* al3 = (const float*)d_in[12];
  const float* ar3 = (const float*)d_in[13];
  const float* b3  = (const float*)d_in[14];

  float* ws   = (float*)d_ws;
  const int HF = H * D;                       // 256
  float* feat = ws;                           // [N, 256]
  float* hbuf = feat + (size_t)N * HF;        // [N, 256]
  float* el   = hbuf + (size_t)N * HF;        // [N, 4]
  float* er   = el + (size_t)N * H;           // [N, 4]
  float* m    = er + (size_t)N * H;           // [N, 4]
  float* s    = m  + (size_t)N * H;           // [N, 4]
  float* ex   = s  + (size_t)N * H;           // [E, 4]

  // Layer 1: [N,128] -> [N,4,64], relu
  run_gat_layer(features, IN, W1, al1, ar1, b1, H, D, src, dst, N, E,
                feat, el, er, m, s, ex, hbuf, /*relu=*/1, stream);
  // Layer 2: [N,256] -> [N,4,64], relu (hbuf consumed by GEMM before overwrite)
  run_gat_layer(hbuf, HF, W2, al2, ar2, b2, H, D, src, dst, N, E,
                feat, el, er, m, s, ex, hbuf, /*relu=*/1, stream);
  // Layer 3: [N,256] -> [N,1,64], no relu; mean over H=1 head is identity
  run_gat_layer(hbuf, HF, W3, al3, ar3, b3, /*H=*/1, /*D=*/OUT, src, dst, N, E,
                feat, el, er, m, s, ex, (float*)d_out, /*relu=*/0, stream);
}